// ModelNew_14061722927747
// MI455X (gfx1250) — compile-verified
//
#include <hip/hip_runtime.h>

typedef __attribute__((ext_vector_type(2))) float v2f;
typedef __attribute__((ext_vector_type(8))) float v8f;

// Fold ((conv + cb)*gw + gb)*sc  ==  conv*alpha + beta
__global__ void prep_affine(const float* __restrict__ cb, const float* __restrict__ gw,
                            const float* __restrict__ gb, const float* __restrict__ sc,
                            float* __restrict__ ab) {
  const int c = threadIdx.x;
  if (c < 64) {
    const float a = gw[c] * sc[c];
    ab[c]      = a;                          // alpha
    ab[64 + c] = cb[c] * a + gb[c] * sc[c];  // beta
  }
}

// One wave: 16 Cout x 16 cols x 4 conv rows -> 16 Cout x 4 pooled cols x 1 pooled row.
// Block = 4 waves covering Cout 0..63 over one shared LDS im2col tile.
//
// LDS im2col layout: xim[r][p][n][j], r=conv row (4), p=k/2 (36), n=col (16), j=k&1 (2).
// B element (k, r, n) = x[b][k/9][4*prow + r + (k%9)/3][col0 + n + k%3].
// WMMA B operand (K-base k0+2*half even) == one ds_load_b64:
//   dword addr = (r*36 + 2*s)*32  +  (32*half + 2*l15)   [imm]      [per-lane base]
__global__ __launch_bounds__(128, 1)
void fused_conv_affine_pool(const float* __restrict__ x, const float* __restrict__ w,
                            const float* __restrict__ alpha, const float* __restrict__ beta,
                            float* __restrict__ out) {
  __shared__ float xim[4 * 36 * 16 * 2];  // 4608 floats = 18.4 KB

  const int tid  = threadIdx.x;
  const int lane = tid & 31;
  const int wave = tid >> 5;
  const int half = lane >> 4;   // selects K pair (A/B) and M+8 (C/D)
  const int l15  = lane & 15;   // M for A, N for B

  const int col0 = blockIdx.x << 4;  // conv col base
  const int prow = blockIdx.y;       // pooled row -> conv rows 4*prow..4*prow+3
  const int b    = blockIdx.z;

  // ---- Cooperative im2col stage into LDS (no runtime divides) ----
  {
    const int n = tid & 15;        // col within tile
    const int q = tid >> 4;        // q == ci (8 groups of 16 threads)
    const float* xg = x + (((b * 8 + q) * 130 + (prow << 2)) * 130) + col0 + n;
    #pragma unroll
    for (int t = 0; t < 9; ++t) {  // k = 9q + t  ->  kh = t/3, kw = t%3 (constants)
      const int k = 9 * q + t;
      const int p = k >> 1;
      const int jj = k & 1;
      #pragma unroll
      for (int r = 0; r < 4; ++r) {
        xim[((r * 36 + p) * 16 + n) * 2 + jj] = xg[((t / 3) + r) * 130 + (t % 3)];
      }
    }
  }

  // ---- Preload all A (weight) pairs into registers: one up-front VMEM clause ----
  const int m0 = wave << 4;  // Cout tile base (0,16,32,48)
  const float* pa = w + (m0 + l15) * 72 + 2 * half;
  v2f areg[18];
  #pragma unroll
  for (int s = 0; s < 18; ++s) areg[s] = *(const v2f*)(pa + 4 * s);

  __syncthreads();

  // ---- K loop: 18 steps x 4 rows; each B = one ds_load_b64 (imm offset) ----
  const float* pb = xim + 32 * half + 2 * l15;  // per-lane base
  v8f acc[4] = {};
  #pragma unroll
  for (int s = 0; s < 18; ++s) {
    #pragma unroll
    for (int r = 0; r < 4; ++r) {
      const v2f bb = *(const v2f*)(pb + (r * 36 + 2 * s) * 32);
      acc[r] = __builtin_amdgcn_wmma_f32_16x16x4_f32(
          /*neg_a=*/false, areg[s], /*neg_b=*/false, bb,
          /*c_mod=*/(short)0, acc[r], /*reuse_a=*/false, /*reuse_b=*/false);
    }
  }

  // ---- Affine (scale may be negative: before max), pool rows in-register,
  //      pool cols across lanes, clamp, store ----
  const int cbase = m0 + 8 * half;  // C/D: channel = m0 + v + 8*half
  #pragma unroll
  for (int v = 0; v < 8; ++v) {
    const int c  = cbase + v;
    const float al = alpha[c];
    const float be = beta[c];
    float t = fmaf(al, acc[0][v], be);
    t = fmaxf(t, fmaf(al, acc[1][v], be));
    t = fmaxf(t, fmaf(al, acc[2][v], be));
    t = fmaxf(t, fmaf(al, acc[3][v], be));
    // N = l15: 4 consecutive lanes = 4 consecutive conv cols (both halves).
    t = fmaxf(t, __shfl_xor(t, 1, 32));
    t = fmaxf(t, __shfl_xor(t, 2, 32));
    t = fminf(fmaxf(t, 0.0f), 1.0f);
    if ((lane & 3) == 0) {
      const int pcol = (blockIdx.x << 2) + (l15 >> 2);
      out[((b * 64 + c) * 32 + prow) * 32 + pcol] = t;
    }
  }
}

extern "C" void kernel_launch(void* const* d_in, const int* in_sizes, int n_in,
                              void* d_out, int out_size, void* d_ws, size_t ws_size,
                              hipStream_t stream) {
  const float* x  = (const float*)d_in[0];  // [128,8,130,130]
  const float* w  = (const float*)d_in[1];  // [64,8,3,3]
  const float* cb = (const float*)d_in[2];  // [64]
  const float* gw = (const float*)d_in[3];  // [64]
  const float* gb = (const float*)d_in[4];  // [64]
  const float* sc = (const float*)d_in[5];  // [64,1,1]
  float* out = (float*)d_out;               // [128,64,32,32]
  float* ab  = (float*)d_ws;                // alpha[64] | beta[64]

  prep_affine<<<1, 64, 0, stream>>>(cb, gw, gb, sc, ab);

  dim3 grid(8 /*col tiles*/, 32 /*pooled rows*/, 128 /*batch*/);
  fused_conv_affine_pool<<<grid, 128, 0, stream>>>(x, w, ab, ab + 64, out);
}